// CoordinatePredictionGCN_29669634081268
// MI455X (gfx1250) — compile-verified
//
#include <hip/hip_runtime.h>
#include <hip/hip_bf16.h>
#include <math.h>

typedef __attribute__((ext_vector_type(2))) float v2f;
typedef __attribute__((ext_vector_type(8))) float v8f;

// ---------------------------------------------------------------------------
// Degree / normalization
// ---------------------------------------------------------------------------
__global__ void k_init_deg(float* __restrict__ deg, int n) {
  int i = blockIdx.x * blockDim.x + threadIdx.x;
  if (i < n) deg[i] = 1.0f;               // self-loop contributes 1 to every node
}

__global__ void k_deg_edges(const long long* __restrict__ ei, float* __restrict__ deg, int E) {
  int e = blockIdx.x * blockDim.x + threadIdx.x;
  if (e < E) atomicAdd(&deg[(int)ei[(size_t)E + e]], 1.0f);
}

__global__ void k_rsqrt_inplace(float* __restrict__ d, int n) {
  int i = blockIdx.x * blockDim.x + threadIdx.x;
  if (i < n) d[i] = rsqrtf(d[i]);         // deg >= 1 always (self-loops)
}

// ---------------------------------------------------------------------------
// Transform: ts = (A @ W^T) * dinv[row]; agg initialized to ts (self-loop term)
// One wave per 16x16 output tile, V_WMMA_F32_16X16X4_F32 over K.
//  A layout (32-bit, 16x4): lanes 0-15 -> M=0..15, K={0(v0),1(v1)};
//                           lanes 16-31 -> M=0..15, K={2(v0),3(v1)}.
//  B = W^T (4x16): same K split across lane halves, N = lane&15.
//  C/D (16x16 f32): vgpr i, lanes 0-15 -> M=i, lanes 16-31 -> M=i+8; N = lane&15.
// ---------------------------------------------------------------------------
template <int K>
__global__ void k_transform_wmma(const float* __restrict__ A,
                                 const float* __restrict__ W,
                                 const float* __restrict__ dinv,
                                 float* __restrict__ ts,
                                 float* __restrict__ agg,
                                 int nNodes, int HOUT) {
  const int lane = threadIdx.x & 31;
  const int waveInBlk = threadIdx.x >> 5;
  const int hTiles = HOUT >> 4;
  const int mTiles = (nNodes + 15) >> 4;
  const int tile = blockIdx.x * (blockDim.x >> 5) + waveInBlk;
  if (tile >= mTiles * hTiles) return;     // wave-uniform: EXEC all-ones inside
  const int mt = tile / hTiles;
  const int ht = tile - mt * hTiles;

  const int ll = lane & 15;                // 0..15
  const int lh = lane >> 4;                // 0 or 1
  int arow = mt * 16 + ll;
  if (arow >= nNodes) arow = nNodes - 1;   // clamp (tail tile), stores guarded below
  const int wrow = ht * 16 + ll;

  const float* __restrict__ Ap = A + (size_t)arow * K + lh * 2;
  const float* __restrict__ Wp = W + (size_t)wrow * K + lh * 2;

  v8f c = {};
#pragma unroll
  for (int k0 = 0; k0 < K; k0 += 4) {
    v2f a = *(const v2f*)(Ap + k0);
    v2f b = *(const v2f*)(Wp + k0);
    c = __builtin_amdgcn_wmma_f32_16x16x4_f32(
        /*neg_a=*/false, a, /*neg_b=*/false, b,
        /*c_mod=*/(short)0, c, /*reuse_a=*/false, /*reuse_b=*/false);
  }

  const int col = ht * 16 + ll;
#pragma unroll
  for (int i = 0; i < 8; ++i) {
    const int r = mt * 16 + i + lh * 8;
    if (r < nNodes) {
      const float v = c[i] * dinv[r];
      const size_t idx = (size_t)r * HOUT + col;
      ts[idx] = v;     // gathered by src in scatter pass
      agg[idx] = v;    // self-loop pre-seed for dst accumulation
    }
  }
}

// ---------------------------------------------------------------------------
// Edge scatter: agg[dst] += ts[src]
// One thread per (edge, 4-feature chunk): b128 gather + 4 f32 L2 atomics.
// ts/agg fit in the 192MB L2, so this runs at L2-atomic rate; vectorizing
// the gather cuts index + data load instructions 4x vs scalar form.
// ---------------------------------------------------------------------------
template <int HOUT>
__global__ void k_edge_scatter(const long long* __restrict__ ei,
                               const float4* __restrict__ ts4,
                               float* __restrict__ agg, int E) {
  constexpr int C = HOUT / 4;                       // chunks per edge (16 or 8)
  constexpr int SH = (HOUT == 64) ? 4 : 3;
  const long long gid = (long long)blockIdx.x * blockDim.x + threadIdx.x;
  const int e = (int)(gid >> SH);
  const int c = (int)(gid & (C - 1));
  if (e >= E) return;
  const int s = (int)ei[e];
  const int d = (int)ei[(size_t)E + e];
  const float4 v = ts4[(size_t)s * C + c];
  float* p = agg + (size_t)d * HOUT + c * 4;
  atomicAdd(p + 0, v.x);
  atomicAdd(p + 1, v.y);
  atomicAdd(p + 2, v.z);
  atomicAdd(p + 3, v.w);
}

// ---------------------------------------------------------------------------
// Finalize: h = relu(dinv[n] * agg[n] + bias)   (float4 vectorized)
// ---------------------------------------------------------------------------
template <int HOUT>
__global__ void k_finalize(const float4* __restrict__ agg4,
                           const float* __restrict__ dinv,
                           const float4* __restrict__ b4,
                           float4* __restrict__ h4, int nNodes) {
  constexpr int C = HOUT / 4;
  constexpr int SH = (HOUT == 64) ? 4 : 3;
  const long long gid = (long long)blockIdx.x * blockDim.x + threadIdx.x;
  const int n = (int)(gid >> SH);
  const int c = (int)(gid & (C - 1));
  if (n >= nNodes) return;
  const float s = dinv[n];
  const float4 a = agg4[(size_t)n * C + c];
  const float4 b = b4[c];
  float4 r;
  r.x = fmaxf(fmaf(s, a.x, b.x), 0.0f);
  r.y = fmaxf(fmaf(s, a.y, b.y), 0.0f);
  r.z = fmaxf(fmaf(s, a.z, b.z), 0.0f);
  r.w = fmaxf(fmaf(s, a.w, b.w), 0.0f);
  h4[(size_t)n * C + c] = r;
}

// ---------------------------------------------------------------------------
// MLP heads: coords = relu(h3@Wc1^T+bc1)@Wc2^T+bc2 ; conf = sigmoid(...)
// One thread per node; weights broadcast via scalar/L0 cache.
// ---------------------------------------------------------------------------
__global__ void k_heads(const float* __restrict__ h3,
                        const float* __restrict__ Wc1, const float* __restrict__ bc1,
                        const float* __restrict__ Wc2, const float* __restrict__ bc2,
                        const float* __restrict__ Wf1, const float* __restrict__ bf1,
                        const float* __restrict__ Wf2, const float* __restrict__ bf2,
                        float* __restrict__ out, int nNodes) {
  const int n = blockIdx.x * blockDim.x + threadIdx.x;
  if (n >= nNodes) return;

  float r[32];
#pragma unroll
  for (int k = 0; k < 32; k += 4) {
    const float4 v = *(const float4*)(h3 + (size_t)n * 32 + k);
    r[k] = v.x; r[k + 1] = v.y; r[k + 2] = v.z; r[k + 3] = v.w;
  }

  float c0 = bc2[0], c1v = bc2[1];
#pragma unroll
  for (int j = 0; j < 16; ++j) {
    float acc = bc1[j];
#pragma unroll
    for (int k = 0; k < 32; ++k) acc = fmaf(Wc1[j * 32 + k], r[k], acc);
    acc = fmaxf(acc, 0.0f);
    c0 = fmaf(Wc2[j], acc, c0);
    c1v = fmaf(Wc2[16 + j], acc, c1v);
  }

  float z = bf2[0];
#pragma unroll
  for (int j = 0; j < 16; ++j) {
    float acc = bf1[j];
#pragma unroll
    for (int k = 0; k < 32; ++k) acc = fmaf(Wf1[j * 32 + k], r[k], acc);
    z = fmaf(Wf2[j], fmaxf(acc, 0.0f), z);
  }

  out[(size_t)n * 2 + 0] = c0;
  out[(size_t)n * 2 + 1] = c1v;
  out[(size_t)2 * nNodes + n] = 1.0f / (1.0f + expf(-z));
}

// ---------------------------------------------------------------------------
extern "C" void kernel_launch(void* const* d_in, const int* in_sizes, int n_in,
                              void* d_out, int out_size, void* d_ws, size_t ws_size,
                              hipStream_t stream) {
  const float*     x   = (const float*)d_in[0];
  const long long* ei  = (const long long*)d_in[1];   // int64 edge_index [2,E]
  const float* W1 = (const float*)d_in[2],  *b1 = (const float*)d_in[3];
  const float* W2 = (const float*)d_in[4],  *b2 = (const float*)d_in[5];
  const float* W3 = (const float*)d_in[6],  *b3 = (const float*)d_in[7];
  const float* Wc1 = (const float*)d_in[8],  *bc1 = (const float*)d_in[9];
  const float* Wc2 = (const float*)d_in[10], *bc2 = (const float*)d_in[11];
  const float* Wf1 = (const float*)d_in[12], *bf1 = (const float*)d_in[13];
  const float* Wf2 = (const float*)d_in[14], *bf2 = (const float*)d_in[15];
  float* out = (float*)d_out;

  const int N = in_sizes[0] / 128;   // 100000
  const int E = in_sizes[1] / 2;     // 1600000
  const int H = 64;

  // workspace carve-out (256B aligned): dinv[N], ts[N*64], agg[N*64], h[N*64]
  char* ws = (char*)d_ws;
  size_t off = 0;
  auto carve = [&](size_t bytes) -> float* {
    float* p = (float*)(ws + off);
    off = (off + bytes + 255) & ~(size_t)255;
    return p;
  };
  float* dinv = carve((size_t)N * sizeof(float));
  float* ts   = carve((size_t)N * H * sizeof(float));
  float* agg  = carve((size_t)N * H * sizeof(float));
  float* h    = carve((size_t)N * H * sizeof(float));
  (void)ws_size; (void)n_in; (void)out_size;

  const int T = 256;
  // ---- symmetric normalization ------------------------------------------
  k_init_deg<<<(N + T - 1) / T, T, 0, stream>>>(dinv, N);
  k_deg_edges<<<(E + T - 1) / T, T, 0, stream>>>(ei, dinv, E);
  k_rsqrt_inplace<<<(N + T - 1) / T, T, 0, stream>>>(dinv, N);

  const int mT = (N + 15) / 16;
  auto gemmBlocks = [&](int hout) { int waves = mT * (hout / 16); return (waves + 7) / 8; };
  auto chunkBlocks = [&](long long elems) { return (int)((elems + T - 1) / T); };

  // ---- layer 1: 128 -> 64 -----------------------------------------------
  k_transform_wmma<128><<<gemmBlocks(64), T, 0, stream>>>(x, W1, dinv, ts, agg, N, 64);
  k_edge_scatter<64><<<chunkBlocks((long long)E * 16), T, 0, stream>>>(ei, (const float4*)ts, agg, E);
  k_finalize<64><<<chunkBlocks((long long)N * 16), T, 0, stream>>>(
      (const float4*)agg, dinv, (const float4*)b1, (float4*)h, N);

  // ---- layer 2: 64 -> 64 ------------------------------------------------
  k_transform_wmma<64><<<gemmBlocks(64), T, 0, stream>>>(h, W2, dinv, ts, agg, N, 64);
  k_edge_scatter<64><<<chunkBlocks((long long)E * 16), T, 0, stream>>>(ei, (const float4*)ts, agg, E);
  k_finalize<64><<<chunkBlocks((long long)N * 16), T, 0, stream>>>(
      (const float4*)agg, dinv, (const float4*)b2, (float4*)h, N);

  // ---- layer 3: 64 -> 32 ------------------------------------------------
  k_transform_wmma<64><<<gemmBlocks(32), T, 0, stream>>>(h, W3, dinv, ts, agg, N, 32);
  k_edge_scatter<32><<<chunkBlocks((long long)E * 8), T, 0, stream>>>(ei, (const float4*)ts, agg, E);
  k_finalize<32><<<chunkBlocks((long long)N * 8), T, 0, stream>>>(
      (const float4*)agg, dinv, (const float4*)b3, (float4*)h, N);

  // ---- heads -------------------------------------------------------------
  k_heads<<<(N + T - 1) / T, T, 0, stream>>>(h, Wc1, bc1, Wc2, bc2, Wf1, bf1, Wf2, bf2, out, N);
}